// GVAE_70291434766876
// MI455X (gfx1250) — compile-verified
//
#include <hip/hip_runtime.h>

// Graph VAE forward for MI455X (gfx1250, wave32).
// - Degrees computed once (graph-only), rsqrt in place.
// - SpMM: wave-per-edge coalesced f32 atomic scatter, deg_out^-1/2 folded into the read.
// - GEMMs: V_WMMA_F32_16X16X4_F32, persistent blocks, W panel staged in LDS with
//   k-pair interleaving so each B fragment is one ds_load_b64 into an even VGPR pair.
//   deg_in^-1/2 folded into the A fragment; bias/ReLU fused in the epilogue.

typedef __attribute__((ext_vector_type(2))) float v2f;
typedef __attribute__((ext_vector_type(8))) float v8f;

#define NN 100000
#define EE 1600000
#define DD 128
#define HH 256
#define ZZ 64

// ---------------- degree kernels ----------------
__global__ void deg_kernel(const int* __restrict__ src, const int* __restrict__ dst,
                           float* __restrict__ degO, float* __restrict__ degI, int E) {
    int e = blockIdx.x * blockDim.x + threadIdx.x;
    if (e >= E) return;
    atomicAdd(&degO[src[e]], 1.0f);
    atomicAdd(&degI[dst[e]], 1.0f);
}

__global__ void rsqrt_kernel(float* __restrict__ degO, float* __restrict__ degI, int n) {
    int i = blockIdx.x * blockDim.x + threadIdx.x;
    if (i >= n) return;
    degO[i] = 1.0f / sqrtf(fmaxf(degO[i], 1.0f));
    degI[i] = 1.0f / sqrtf(fmaxf(degI[i], 1.0f));
}

// ---------------- edge scatter: one wave32 per edge, coalesced atomics ----------------
// agg[dst] += x[src] * dinvO[src]   (deg_out^-1/2 folded in; multiply is free vs atomic)
__global__ __launch_bounds__(256) void scatter_kernel(
        const int* __restrict__ src, const int* __restrict__ dst,
        const float* __restrict__ x, const float* __restrict__ dinvO,
        float* __restrict__ agg, int E, int w) {
    int wave = (blockIdx.x * blockDim.x + threadIdx.x) >> 5;
    int lane = threadIdx.x & 31;
    if (wave >= E) return;
    int s = src[wave];
    int d = dst[wave];
    float sc = dinvO[s];
    const float* __restrict__ xrow = x + (size_t)s * w;
    float* __restrict__ arow = agg + (size_t)d * w;
    for (int j = lane; j < w; j += 32)
        atomicAdd(&arow[j], xrow[j] * sc);
}

// ---------------- WMMA f32 GEMM: C = (A * dinvI[row]) @ W + bias ----------------
// grid = (Nc/64, NBY); block = 256 (8 waves). Each block owns one 64-column panel of W,
// stages it to LDS once (k-pair interleaved), then its 8 waves sweep M-tiles persistently.
// LDS layout: smem[(k>>1)*128 + 2*c + (k&1)] = W[k][tg*64+c]  ->  B frag {W[k][c],W[k+1][c]}
// is a single 8B-aligned ds_load_b64. Each wave: 16x64 strip = 4 accumulators per A frag.
// MODE 0: bias add.  MODE 1: bias + ReLU.
template <int MODE>
__global__ __launch_bounds__(256) void gemm_wmma_kernel(
        const float* __restrict__ A, const float* __restrict__ dinvI,
        const float* __restrict__ W, const float* __restrict__ bias,
        float* __restrict__ C, int M, int K, int Nc) {
    extern __shared__ float smem[];                      // K x 64 panel, k-pair interleaved
    const int tid  = threadIdx.x;
    const int lane = tid & 31;
    const int wid  = tid >> 5;
    const int tg   = blockIdx.x;                         // 64-column group

    // cooperative stage: float4 global reads, interleaved scalar LDS writes
    for (int t = tid; t < (K << 4); t += 256) {          // K*16 float4s
        int k  = t >> 4;
        int c4 = (t & 15) << 2;
        float4 v = *((const float4*)(W + (size_t)k * Nc + (tg << 6) + c4));
        float* dp = smem + ((k >> 1) << 7) + (c4 << 1) + (k & 1);
        dp[0] = v.x; dp[2] = v.y; dp[4] = v.z; dp[6] = v.w;
    }
    __syncthreads();

    const int Mtiles = M >> 4;
    const int half = lane >> 4;                          // K-pair select: {0,1} vs {2,3}
    const int l15  = lane & 15;
    const v2f* __restrict__ bbase = (const v2f*)smem + l15;   // v2f units: pair p, col c

    for (int tm = blockIdx.y * 8 + wid; tm < Mtiles; tm += gridDim.y * 8) {
        const int m0   = tm << 4;
        const int arow = m0 + l15;
        const float ds = dinvI[arow];                    // fold deg_in^-1/2 into A
        const float* __restrict__ ap = A + (size_t)arow * K + half * 2;

        v8f c[4] = {};
        for (int k0 = 0; k0 < K; k0 += 4) {
            v2f a;
            a.x = ap[0] * ds;
            a.y = ap[1] * ds;
            ap += 4;
            const v2f* __restrict__ bs = bbase + (((k0 >> 1) + half) << 6);
            v2f bf[4];
#pragma unroll
            for (int nt = 0; nt < 4; ++nt)               // 4x ds_load_b64, no shuffles
                bf[nt] = bs[nt * 16];
#pragma unroll
            for (int nt = 0; nt < 4; ++nt)
                c[nt] = __builtin_amdgcn_wmma_f32_16x16x4_f32(
                            false, a, false, bf[nt], (short)0, c[nt], false, false);
        }

        const int n0 = (tg << 6) + l15;
#pragma unroll
        for (int nt = 0; nt < 4; ++nt) {
            const int col = n0 + nt * 16;
            const float bv = bias[col];
#pragma unroll
            for (int r = 0; r < 8; ++r) {                // 16x16 f32 C/D layout
                float v = c[nt][r] + bv;
                if (MODE == 1) v = fmaxf(v, 0.0f);
                C[(size_t)(m0 + r + half * 8) * Nc + col] = v;
            }
        }
    }
}

// ---------------- split mean/logvar + reparameterize ----------------
__global__ void split_z_kernel(const float* __restrict__ ml, const float* __restrict__ eps,
                               float* __restrict__ meanO, float* __restrict__ lvO,
                               float* __restrict__ z, int n) {
    int t = blockIdx.x * blockDim.x + threadIdx.x;
    if (t >= n * ZZ) return;
    int row = t >> 6;
    int j   = t & (ZZ - 1);
    float mn = ml[(size_t)row * DD + j];
    float lv = ml[(size_t)row * DD + ZZ + j];
    meanO[t] = mn;
    lvO[t]   = lv;
    z[t] = eps[t] * expf(0.5f * lv) + mn;
}

extern "C" void kernel_launch(void* const* d_in, const int* in_sizes, int n_in,
                              void* d_out, int out_size, void* d_ws, size_t ws_size,
                              hipStream_t stream) {
    const float* feat = (const float*)d_in[0];
    const int*   src  = (const int*)d_in[1];
    const int*   dst  = (const int*)d_in[2];
    const float* eps  = (const float*)d_in[3];
    const float* W1   = (const float*)d_in[4];
    const float* b1   = (const float*)d_in[5];
    const float* W2   = (const float*)d_in[6];
    const float* b2   = (const float*)d_in[7];
    const float* W3   = (const float*)d_in[8];
    const float* b3   = (const float*)d_in[9];

    float* out   = (float*)d_out;
    float* recon = out;                               // N*128
    float* meanO = out + (size_t)NN * DD;             // N*64
    float* lvO   = meanO + (size_t)NN * ZZ;           // N*64

    float* ws    = (float*)d_ws;
    float* dinvO = ws;                                // N  (deg -> rsqrt in place)
    float* dinvI = ws + NN;                           // N
    float* buf0  = ws + 2 * (size_t)NN;               // N*256 (h, then ml/z reuse)
    float* buf1  = buf0 + (size_t)NN * HH;            // N*256 (agg)
    float* ml    = buf0;                              // N*128 (reuse after scatter2)
    float* z     = buf0 + (size_t)NN * DD;            // N*64  (reuse)

    const int T   = 256;
    const int NBY = 200;                              // persistent M-sweep depth

    // degrees (graph-only)
    hipMemsetAsync(dinvO, 0, 2ull * NN * sizeof(float), stream);
    deg_kernel<<<(EE + T - 1) / T, T, 0, stream>>>(src, dst, dinvO, dinvI, EE);
    rsqrt_kernel<<<(NN + T - 1) / T, T, 0, stream>>>(dinvO, dinvI, NN);

    // ---- layer 1: scatter(x * dO) -> WMMA GEMM (ReLU) -> h ----
    hipMemsetAsync(buf1, 0, (size_t)NN * DD * sizeof(float), stream);
    scatter_kernel<<<(EE * 32 + T - 1) / T, T, 0, stream>>>(src, dst, feat, dinvO, buf1, EE, DD);
    gemm_wmma_kernel<1><<<dim3(HH / 64, NBY), T, DD * 64 * sizeof(float), stream>>>(
        buf1, dinvI, W1, b1, buf0, NN, DD, HH);

    // ---- layer 2: scatter(h * dO) -> WMMA GEMM -> mean|logvar -> z ----
    hipMemsetAsync(buf1, 0, (size_t)NN * HH * sizeof(float), stream);
    scatter_kernel<<<(EE * 32 + T - 1) / T, T, 0, stream>>>(src, dst, buf0, dinvO, buf1, EE, HH);
    gemm_wmma_kernel<0><<<dim3(DD / 64, NBY), T, HH * 64 * sizeof(float), stream>>>(
        buf1, dinvI, W2, b2, ml, NN, HH, DD);
    split_z_kernel<<<(NN * ZZ + T - 1) / T, T, 0, stream>>>(ml, eps, meanO, lvO, z, NN);

    // ---- layer 3: scatter(z * dO) -> WMMA GEMM -> recon ----
    hipMemsetAsync(buf1, 0, (size_t)NN * ZZ * sizeof(float), stream);
    scatter_kernel<<<(EE * 32 + T - 1) / T, T, 0, stream>>>(src, dst, z, dinvO, buf1, EE, ZZ);
    gemm_wmma_kernel<0><<<dim3(DD / 64, NBY), T, ZZ * 64 * sizeof(float), stream>>>(
        buf1, dinvI, W3, b3, recon, NN, ZZ, DD);
}